// rotConvLayer_64518998720857
// MI455X (gfx1250) — compile-verified
//
#include <hip/hip_runtime.h>
#include <hip/hip_bf16.h>
#include <math.h>

// ---------------------------------------------------------------------------
// rot-conv for MI455X (gfx1250), wave32 + V_WMMA_F32_16X16X4_F32 implicit GEMM
//   x: (32, 3, 224, 224) f32, w: (24, 1, 3, 3) f32
//   out: (32, 192, 222, 222) f32   (~1.21 GB -> store-bandwidth bound)
// Round 2: LDS-staged output tile -> coalesced 256B-per-wave b64 stores.
// ---------------------------------------------------------------------------

typedef __attribute__((ext_vector_type(2))) float v2f;
typedef __attribute__((ext_vector_type(8))) float v8f;

#define MM 3
#define NN 8
#define GG 8
#define KK 3
#define IH 224
#define IW 224
#define OH 222
#define OW 222
#define OC 192          // M*N*G
#define KPAD 12         // 9 taps padded to 12 (3 WMMA K-steps of 4)
#define TILE_X 128      // 8 waves * 16 px
#define LDS_PITCH 132   // padded pitch: 132 % 64 = 4 -> conflict-free scatter

// ---------------------------------------------------------------------------
// Kernel 1: build rotated filter bank wbuf[192][12] (taps 9..11 = 0).
// Replicates _rotation_grids() + _grid_sample() exactly (zero-pad bilinear).
// ---------------------------------------------------------------------------
__global__ void rot_weights_kernel(const float* __restrict__ w,
                                   float* __restrict__ wbuf) {
    int o = threadIdx.x;
    if (o >= OC) return;
    int f = o >> 3;          // 0..23  (filter = m*8+n)
    int g = o & 7;           // 0..7   (rotation)

    const float PI = 3.14159265358979323846f;
    float ang = (-90.0f + (float)g * (180.0f / 7.0f)) * (PI / 180.0f);
    float c = cosf(ang), s = sinf(ang);

    const float* wf = w + f * (KK * KK);

    #pragma unroll
    for (int h = 0; h < KK; ++h) {
        float yb = (2.0f * (float)h + 1.0f) / (float)KK - 1.0f;   // ys
        #pragma unroll
        for (int wcol = 0; wcol < KK; ++wcol) {
            float xb = (2.0f * (float)wcol + 1.0f) / (float)KK - 1.0f; // xs
            // grid = theta * (xb, yb, 1)
            float gx = c * xb - s * yb;
            float gy = s * xb + c * yb;
            // bilinear sample coords
            float ix = ((gx + 1.0f) * (float)KK - 1.0f) * 0.5f;
            float iy = ((gy + 1.0f) * (float)KK - 1.0f) * 0.5f;
            float x0f = floorf(ix), y0f = floorf(iy);
            float wx1 = ix - x0f, wy1 = iy - y0f;
            float wx0 = 1.0f - wx1, wy0 = 1.0f - wy1;
            float acc = 0.0f;
            #pragma unroll
            for (int dy = 0; dy < 2; ++dy) {
                #pragma unroll
                for (int dx = 0; dx < 2; ++dx) {
                    float xf = x0f + (float)dx;
                    float yf = y0f + (float)dy;
                    float valid = (xf >= 0.0f && xf < (float)KK &&
                                   yf >= 0.0f && yf < (float)KK) ? 1.0f : 0.0f;
                    int xc = (int)fminf(fmaxf(xf, 0.0f), (float)(KK - 1));
                    int yc = (int)fminf(fmaxf(yf, 0.0f), (float)(KK - 1));
                    float ww = (dx ? wx1 : wx0) * (dy ? wy1 : wy0);
                    acc += wf[yc * KK + xc] * valid * ww;
                }
            }
            wbuf[o * KPAD + h * KK + wcol] = acc;
        }
    }
    // zero K-padding taps 9..11 so the padded GEMM is exact
    wbuf[o * KPAD + 9]  = 0.0f;
    wbuf[o * KPAD + 10] = 0.0f;
    wbuf[o * KPAD + 11] = 0.0f;
}

// ---------------------------------------------------------------------------
// Kernel 2: implicit GEMM, one block = (b, m, row y, 128-px x tile),
// 8 wave32; each wave: 16 px * 64 oc via 4 accumulators, 12 WMMAs.
// Output staged through LDS for coalesced b64 stores (256B/wave/inst).
// ---------------------------------------------------------------------------
__global__ __launch_bounds__(256) void rot_conv_kernel(
        const float* __restrict__ x,
        const float* __restrict__ wbuf,
        float* __restrict__ out) {
    __shared__ float smem[(3 + 64) * LDS_PITCH];
    float* sin_ = smem;                   // 3 x 132 staged input rows
    float* sout = smem + 3 * LDS_PITCH;   // 64 x 132 output tile

    const int m  = blockIdx.y;          // group / input channel
    const int b  = blockIdx.z;          // batch
    const int y  = blockIdx.x >> 1;     // output row 0..221
    const int x0 = (blockIdx.x & 1) * TILE_X;

    // ---- stage 3 input rows (y..y+2), cols x0..x0+129 into LDS ----
    const float* xin = x + (((size_t)b * MM + m) * IH + y) * IW;
    for (int idx = threadIdx.x; idx < 3 * LDS_PITCH; idx += 256) {
        int r  = idx / LDS_PITCH;
        int i  = idx - r * LDS_PITCH;
        int xi = x0 + i;
        if (xi > IW - 1) xi = IW - 1;          // clamp; clamped data never stored
        sin_[r * LDS_PITCH + i] = xin[r * IW + xi];
    }
    __syncthreads();

    const int lane    = threadIdx.x & 31;
    const int wave    = threadIdx.x >> 5;
    const int halfrow = lane & 15;
    const bool hi     = lane >= 16;

    // ---- A fragments: weights, ISA 16x4 f32 A layout ----
    // lanes 0-15: M=lane,  VGPR0->K=4k+0, VGPR1->K=4k+1
    // lanes 16-31: M=lane-16, VGPR0->K=4k+2, VGPR1->K=4k+3
    v2f A[4][3];
    const float* wb = wbuf + (size_t)m * 64 * KPAD;
    #pragma unroll
    for (int t = 0; t < 4; ++t) {
        #pragma unroll
        for (int k = 0; k < 3; ++k) {
            const float* p = wb + (16 * t + halfrow) * KPAD + 4 * k + (hi ? 2 : 0);
            v2f a;
            a.x = p[0];
            a.y = p[1];
            A[t][k] = a;
        }
    }

    v8f acc[4] = {(v8f)0.0f, (v8f)0.0f, (v8f)0.0f, (v8f)0.0f};

    // ---- K loop: 3 steps of 4 taps; B from LDS (4x16 layout mirrors A) ----
    const float* ldsw = sin_ + wave * 16 + halfrow;
    #pragma unroll
    for (int k = 0; k < 3; ++k) {
        int tap0 = 4 * k + (hi ? 2 : 0);
        int t0 = tap0     > 8 ? 8 : tap0;      // taps 9..11: A is zero, B value moot
        int t1 = tap0 + 1 > 8 ? 8 : tap0 + 1;
        v2f B;
        B.x = ldsw[(t0 / 3) * LDS_PITCH + (t0 % 3)];
        B.y = ldsw[(t1 / 3) * LDS_PITCH + (t1 % 3)];
        #pragma unroll
        for (int t = 0; t < 4; ++t) {
            acc[t] = __builtin_amdgcn_wmma_f32_16x16x4_f32(
                /*neg_a=*/false, A[t][k],
                /*neg_b=*/false, B,
                /*c_mod=*/(short)0, acc[t],
                /*reuse_a=*/false, /*reuse_b=*/false);
        }
    }

    // ---- scatter D into LDS tile per ISA 16x16 f32 C/D layout ----
    // VGPR j: lanes 0-15 -> (M=j, N=lane); lanes 16-31 -> (M=8+j, N=lane-16)
    // Half-wave rows differ by 8*132 % 64 = 32 banks -> conflict-free.
    {
        const int pxl = wave * 16 + halfrow;
        #pragma unroll
        for (int t = 0; t < 4; ++t) {
            int ocb = 16 * t + (hi ? 8 : 0);
            #pragma unroll
            for (int j = 0; j < 8; ++j)
                sout[(ocb + j) * LDS_PITCH + pxl] = acc[t][j];
        }
    }
    __syncthreads();

    // ---- cooperative coalesced store: wave = half an oc row = 256B/inst ----
    // All addresses are 8B-aligned: row base (r*222 + x0) and px are even.
    const int vw = (OW - x0 < TILE_X) ? (OW - x0) : TILE_X;   // valid width
    float* obase = out + (((size_t)b * OC + m * 64) * OH + y) * OW + x0;
    #pragma unroll
    for (int i = 0; i < 16; ++i) {
        int idx = (int)threadIdx.x + i * 256;   // 0..4095
        int oc  = idx >> 6;                     // 0..63
        int px  = (idx & 63) * 2;               // 0..126, even
        float2 v = *(const float2*)&sout[oc * LDS_PITCH + px];
        float* dst = obase + (size_t)oc * (OH * OW) + px;
        if (px + 2 <= vw) {
            *(float2*)dst = v;                  // global_store_b64, aligned
        } else if (px < vw) {
            dst[0] = v.x;                       // odd-width edge (unused here)
        }
    }
}

// ---------------------------------------------------------------------------
extern "C" void kernel_launch(void* const* d_in, const int* in_sizes, int n_in,
                              void* d_out, int out_size, void* d_ws, size_t ws_size,
                              hipStream_t stream) {
    const float* x = (const float*)d_in[0];   // (32,3,224,224)
    const float* w = (const float*)d_in[1];   // (24,1,3,3)
    float* out  = (float*)d_out;              // (32,192,222,222)
    float* wbuf = (float*)d_ws;               // 192*12 floats staging

    rot_weights_kernel<<<1, OC, 0, stream>>>(w, wbuf);

    dim3 grid(OH * 2, MM, 32);                // (y * xtiles, group, batch)
    rot_conv_kernel<<<grid, 256, 0, stream>>>(x, wbuf, out);
}